// MedianEncoder_75814762709162
// MI455X (gfx1250) — compile-verified
//
#include <hip/hip_runtime.h>
#include <math.h>

#define N_NODES 50000
#define N_EDGES 800000
#define F_DIM   128
#define H_DIM   64
#define K_CL    16
#define MAXD    192   // Poisson(16) max over 50k nodes is ~50; generous cap

typedef float    v2f   __attribute__((ext_vector_type(2)));
typedef float    v8f   __attribute__((ext_vector_type(8)));
typedef unsigned u32x4 __attribute__((ext_vector_type(4)));
typedef int      i32x4 __attribute__((ext_vector_type(4)));
typedef int      i32x8 __attribute__((ext_vector_type(8)));

// ---------------- Tensor Data Mover: 1-D tile of n ints, global -> LDS ----------------
// D# per CDNA5 ISA: group0 = {count=1, lds_addr, global_addr,
// type=2}; group1 = {data_size=4B, tensor_dim0=n, tensor_dim1=1, tile_dim0=n,
// tile_dim1=1, tensor_dim0_stride=n}; groups 2/3 zero (<=2-D tensor, unused dims).
// All fields are wave-uniform here. tile_dim0==0 => NOP (empty segment).
__device__ __forceinline__ void tdm_load_ints_to_lds(unsigned lds_byte,
                                                     const int* gptr, int n) {
    unsigned long long ga = (unsigned long long)(uintptr_t)gptr;
    u32x4 g0 = { 1u,                                   // count=1, user descriptor
                 lds_byte,                             // lds_addr (bytes)
                 (unsigned)(ga & 0xFFFFFFFFu),         // global_addr[31:0]
                 (unsigned)((ga >> 32) & 0x1FFFFFFu) | (2u << 30) }; // [56:32] | type=2
    unsigned un = (unsigned)n;
    i32x8 g1 = { (int)(2u << 16),        // workgroup_mask=0, data_size=2 (4 bytes)
                 (int)(un << 16),        // tensor_dim0[15:0] -> bits63:48
                 (int)((un >> 16) | (1u << 16)), // tensor_dim0[31:16] | tensor_dim1=1
                 (int)(un << 16),        // tile_dim0 = n (bits127:112)
                 1,                      // tile_dim1=1, tile_dim2=0
                 (int)un,                // tensor_dim0_stride[31:0] = n
                 0, 0 };
    i32x4 z4 = { 0, 0, 0, 0 };
#if defined(__clang_major__) && (__clang_major__ >= 23)
    i32x8 z8 = { 0, 0, 0, 0, 0, 0, 0, 0 };
    __builtin_amdgcn_tensor_load_to_lds(g0, g1, z4, z4, z8, 0);
#else
    __builtin_amdgcn_tensor_load_to_lds(g0, g1, z4, z4, 0);
#endif
}

// ---------------- CSR construction ----------------

__global__ void zero_ints(int* __restrict__ p, int n) {
    int i = blockIdx.x * blockDim.x + threadIdx.x;
    if (i < n) p[i] = 0;
}

__global__ void count_edges(const int* __restrict__ dst, int* __restrict__ deg, int e) {
    int i = blockIdx.x * blockDim.x + threadIdx.x;
    if (i < e) atomicAdd(&deg[dst[i]], 1);
}

// single-workgroup chunked Hillis-Steele exclusive scan (50k elems, 49 chunks)
__global__ void scan_excl(const int* __restrict__ deg, int* __restrict__ start, int n) {
    __shared__ int buf[1024];
    __shared__ int carry;
    if (threadIdx.x == 0) carry = 0;
    __syncthreads();
    for (int base = 0; base < n; base += 1024) {
        int i = base + (int)threadIdx.x;
        int v = (i < n) ? deg[i] : 0;
        buf[threadIdx.x] = v;
        __syncthreads();
        for (int off = 1; off < 1024; off <<= 1) {
            int t = (threadIdx.x >= (unsigned)off) ? buf[threadIdx.x - off] : 0;
            __syncthreads();
            buf[threadIdx.x] += t;
            __syncthreads();
        }
        if (i < n) start[i] = carry + buf[threadIdx.x] - v;   // exclusive
        __syncthreads();
        if (threadIdx.x == 1023) carry += buf[1023];
        __syncthreads();
    }
    if (threadIdx.x == 0) start[n] = carry;
}

__global__ void scatter_edges(const int* __restrict__ src, const int* __restrict__ dst,
                              const int* __restrict__ start, int* __restrict__ cursor,
                              int* __restrict__ csr_src, int e) {
    int i = blockIdx.x * blockDim.x + threadIdx.x;
    if (i < e) {
        int d   = dst[i];
        int pos = start[d] + atomicAdd(&cursor[d], 1);
        csr_src[pos] = src[i];
    }
}

// ---------------- GEMMs via V_WMMA_F32_16X16X4_F32 ----------------
// One wave = one 16x16 output tile. A/B fragments are float2 loads matching
// the ISA 16x4 / 4x16 f32 layouts (lane m/n = lane%16, K pair = 2*(lane/16)).

__global__ void gemm1_wmma(const float* __restrict__ X, const float* __restrict__ W1,
                           const float* __restrict__ b1, float* __restrict__ hlin) {
    int wave = threadIdx.x >> 5;
    int lane = threadIdx.x & 31;
    int tile = blockIdx.x * 8 + wave;
    if (tile >= (N_NODES / 16) * (H_DIM / 16)) return;
    int mt = tile >> 2, nt = tile & 3;        // 4 N-tiles (H=64)
    int m0 = mt * 16, n0 = nt * 16;
    int l16 = lane & 15, hi = lane >> 4;
    const float* arow = X  + (size_t)(m0 + l16) * F_DIM + hi * 2;
    const float* brow = W1 + (size_t)(n0 + l16) * F_DIM + hi * 2;   // B = W1^T via row reads
    v8f acc = {0.f, 0.f, 0.f, 0.f, 0.f, 0.f, 0.f, 0.f};
#pragma unroll
    for (int kk = 0; kk < F_DIM; kk += 4) {
        v2f a = *(const v2f*)(arow + kk);
        v2f b = *(const v2f*)(brow + kk);
        acc = __builtin_amdgcn_wmma_f32_16x16x4_f32(false, a, false, b,
                                                    (short)0, acc, false, false);
    }
    float bias = b1[n0 + l16];
    int rbase = m0 + hi * 8;
#pragma unroll
    for (int r = 0; r < 8; ++r)
        hlin[(size_t)(rbase + r) * H_DIM + (n0 + l16)] = acc[r] + bias;
}

__global__ void gemm2_wmma(const float* __restrict__ Hact, const float* __restrict__ W2,
                           const float* __restrict__ b2, float* __restrict__ lg) {
    int wave = threadIdx.x >> 5;
    int lane = threadIdx.x & 31;
    int tile = blockIdx.x * 8 + wave;
    if (tile >= N_NODES / 16) return;          // N dim = 16 -> single n-tile
    int m0 = tile * 16;
    int l16 = lane & 15, hi = lane >> 4;
    const float* arow = Hact + (size_t)(m0 + l16) * H_DIM + hi * 2;
    const float* brow = W2   + (size_t)l16 * H_DIM + hi * 2;
    v8f acc = {0.f, 0.f, 0.f, 0.f, 0.f, 0.f, 0.f, 0.f};
#pragma unroll
    for (int kk = 0; kk < H_DIM; kk += 4) {
        v2f a = *(const v2f*)(arow + kk);
        v2f b = *(const v2f*)(brow + kk);
        acc = __builtin_amdgcn_wmma_f32_16x16x4_f32(false, a, false, b,
                                                    (short)0, acc, false, false);
    }
    float bias = b2[l16];
    int rbase = m0 + hi * 8;
#pragma unroll
    for (int r = 0; r < 8; ++r)
        lg[(size_t)(rbase + r) * K_CL + l16] = acc[r] + bias;
}

// ---------------- median aggregation (rank-count, L2-resident gathers) ----------------
// Lower median == element with stable rank (d-1)/2:
//   rank(i) = #{v_j < v_i} + #{v_j == v_i && j < i}
// CSR slices are staged into LDS with the Tensor Data Mover (TENSORcnt);
// message rows are warmed with global_prefetch before the O(d^2) rank loops.

__global__ void median1_elu(const int* __restrict__ start, const int* __restrict__ csr_src,
                            const float* __restrict__ hlin, float* __restrict__ hact) {
    __shared__ int sid[8][MAXD];                      // single LDS object -> offset 0
    int wave = threadIdx.x >> 5;
    int lane = threadIdx.x & 31;
    int node = blockIdx.x * 8 + wave;                 // 6250 * 8 == 50000 exactly
    int s = start[node];
    int d = start[node + 1] - s;
    if (d > MAXD) d = MAXD;
    tdm_load_ints_to_lds((unsigned)(wave * MAXD * 4), csr_src + s, d);
    __builtin_amdgcn_s_wait_tensorcnt(0);
    __syncthreads();
    // warm L2->WGP caches for this node's message rows (64 floats = 2 cachelines)
    for (int i = lane; i < d; i += 32) {
        const float* row = hlin + (size_t)sid[wave][i] * H_DIM;
        __builtin_prefetch(row, 0, 3);
        __builtin_prefetch(row + 32, 0, 3);
    }
    int target = (d - 1) >> 1;
#pragma unroll
    for (int t = 0; t < 2; ++t) {                     // 64 features / 32 lanes
        int f = lane + t * 32;
        float med = 0.f;
        for (int i = 0; i < d; ++i) {
            float vi = hlin[(size_t)sid[wave][i] * H_DIM + f];
            int rank = 0;
            for (int j = 0; j < d; ++j) {
                float vj = hlin[(size_t)sid[wave][j] * H_DIM + f];
                rank += (vj < vi) || (vj == vi && j < i);
            }
            if (rank == target) med = vi;
        }
        float out = (med > 0.f) ? med : (expf(med) - 1.f);   // ELU, alpha=1
        hact[(size_t)node * H_DIM + f] = out;
    }
}

__global__ void median2_softmax(const int* __restrict__ start, const int* __restrict__ csr_src,
                                const float* __restrict__ lg, float* __restrict__ out) {
    __shared__ int sid[16][MAXD];                     // single LDS object -> offset 0
    int wave  = threadIdx.x >> 5;
    int halfw = (threadIdx.x >> 4) & 1;               // half-wave owns one node
    int l     = threadIdx.x & 15;
    int nodeA = blockIdx.x * 16 + wave * 2;           // 3125 * 16 == 50000 exactly
    int sA = start[nodeA];
    int dA = start[nodeA + 1] - sA;  if (dA > MAXD) dA = MAXD;
    int sB = start[nodeA + 1];
    int dB = start[nodeA + 2] - sB;  if (dB > MAXD) dB = MAXD;
    // two wave-uniform TDM descriptors per wave: one per half-wave's node
    tdm_load_ints_to_lds((unsigned)((wave * 2)     * MAXD * 4), csr_src + sA, dA);
    tdm_load_ints_to_lds((unsigned)((wave * 2 + 1) * MAXD * 4), csr_src + sB, dB);
    __builtin_amdgcn_s_wait_tensorcnt(0);
    __syncthreads();
    int row  = wave * 2 + halfw;
    int node = nodeA + halfw;
    int d    = halfw ? dB : dA;
    for (int i = l; i < d; i += 16)                   // logits rows: 16 floats = 64B
        __builtin_prefetch(lg + (size_t)sid[row][i] * K_CL, 0, 3);
    int target = (d - 1) >> 1;
    float med = 0.f;
    for (int i = 0; i < d; ++i) {
        float vi = lg[(size_t)sid[row][i] * K_CL + l];
        int rank = 0;
        for (int j = 0; j < d; ++j) {
            float vj = lg[(size_t)sid[row][j] * K_CL + l];
            rank += (vj < vi) || (vj == vi && j < i);
        }
        if (rank == target) med = vi;
    }
    // softmax over the 16 lanes of this half-wave (wave32-native shuffles)
    float m = med;
    for (int off = 8; off > 0; off >>= 1) m = fmaxf(m, __shfl_xor(m, off, 16));
    float e = expf(med - m);
    float sum = e;
    for (int off = 8; off > 0; off >>= 1) sum += __shfl_xor(sum, off, 16);
    out[(size_t)node * K_CL + l] = e / sum;
}

// ---------------- launch ----------------

extern "C" void kernel_launch(void* const* d_in, const int* in_sizes, int n_in,
                              void* d_out, int out_size, void* d_ws, size_t ws_size,
                              hipStream_t stream) {
    const float* X  = (const float*)d_in[0];
    const int*   ei = (const int*)d_in[1];      // [2, E]
    const int*  src = ei;
    const int*  dst = ei + N_EDGES;
    const float* W1 = (const float*)d_in[3];
    const float* b1 = (const float*)d_in[4];
    const float* W2 = (const float*)d_in[5];
    const float* b2 = (const float*)d_in[6];
    float* out = (float*)d_out;
    (void)in_sizes; (void)n_in; (void)out_size; (void)ws_size;

    char* ws = (char*)d_ws;
    size_t off = 0;
    auto alloc = [&](size_t bytes) {
        void* p = ws + off;
        off = (off + bytes + 255) & ~(size_t)255;
        return p;
    };
    int*   deg     = (int*)alloc((size_t)N_NODES * 4);
    int*   startp  = (int*)alloc((size_t)(N_NODES + 1) * 4);
    int*   cursor  = (int*)alloc((size_t)N_NODES * 4);
    int*   csr_src = (int*)alloc((size_t)N_EDGES * 4);
    float* hlin    = (float*)alloc((size_t)N_NODES * H_DIM * 4);
    float* hact    = (float*)alloc((size_t)N_NODES * H_DIM * 4);
    float* lgbuf   = (float*)alloc((size_t)N_NODES * K_CL * 4);

    zero_ints<<<(N_NODES + 255) / 256, 256, 0, stream>>>(deg, N_NODES);
    zero_ints<<<(N_NODES + 255) / 256, 256, 0, stream>>>(cursor, N_NODES);
    count_edges<<<(N_EDGES + 255) / 256, 256, 0, stream>>>(dst, deg, N_EDGES);
    scan_excl<<<1, 1024, 0, stream>>>(deg, startp, N_NODES);
    scatter_edges<<<(N_EDGES + 255) / 256, 256, 0, stream>>>(src, dst, startp, cursor,
                                                             csr_src, N_EDGES);
    // layer 1: linear (WMMA f32) -> per-dst median -> ELU
    gemm1_wmma<<<(12500 + 7) / 8, 256, 0, stream>>>(X, W1, b1, hlin);
    median1_elu<<<N_NODES / 8, 256, 0, stream>>>(startp, csr_src, hlin, hact);
    // layer 2: linear (WMMA f32) -> per-dst median -> softmax
    gemm2_wmma<<<(3125 + 7) / 8, 256, 0, stream>>>(hact, W2, b2, lgbuf);
    median2_softmax<<<N_NODES / 16, 256, 0, stream>>>(startp, csr_src, lgbuf, out);
}